// Block_39195871543913
// MI455X (gfx1250) — compile-verified
//
#include <hip/hip_runtime.h>
#include <math.h>

// ---------------------------------------------------------------------------
// Problem constants (from the reference)
// ---------------------------------------------------------------------------
static constexpr int kN = 262144;   // points
static constexpr int kC = 96;       // channels
static constexpr int kK = 49;       // neighbor offsets
static constexpr int kB = 8;        // batches
static constexpr int kH = 384;      // hidden = 4*C
static constexpr float kEpsLN  = 1e-6f;
static constexpr float kEpsGRN = 1e-6f;

typedef __attribute__((ext_vector_type(2))) float v2f;
typedef __attribute__((ext_vector_type(8))) float v8f;

// ---------------------------------------------------------------------------
// Kernel 1: sparse depthwise conv (gather over 49 neighbors) + LayerNorm(96)
// One wave (32 lanes) per point; lane owns channels {l, l+32, l+64}.
// dw_w (49x96 = 18.8KB) staged in LDS. feats (100MB) is L2-resident.
// ---------------------------------------------------------------------------
__global__ __launch_bounds__(256) void dwconv_ln_kernel(
    const float* __restrict__ feats, const float* __restrict__ dw_w,
    const float* __restrict__ dw_b, const float* __restrict__ ln_g,
    const float* __restrict__ ln_b, const int* __restrict__ nbr,
    float* __restrict__ x_ln)
{
    __shared__ float sw[kK * kC];
    for (int i = threadIdx.x; i < kK * kC; i += 256) sw[i] = dw_w[i];
    __syncthreads();

    const int wave = threadIdx.x >> 5;
    const int lane = threadIdx.x & 31;
    const int n = blockIdx.x * 8 + wave;
    const int c0 = lane, c1 = lane + 32, c2 = lane + 64;

    float a0 = 0.f, a1 = 0.f, a2 = 0.f;
    const int* np = nbr + (long)n * kK;
    for (int k = 0; k < kK; ++k) {
        const int idx = np[k];                      // uniform across wave
        const float* fr = feats + (long)idx * kC;   // L2-resident gather
        const float* wr = sw + k * kC;
        a0 = fmaf(fr[c0], wr[c0], a0);
        a1 = fmaf(fr[c1], wr[c1], a1);
        a2 = fmaf(fr[c2], wr[c2], a2);
    }
    a0 += dw_b[c0]; a1 += dw_b[c1]; a2 += dw_b[c2];

    // LayerNorm over the 96 channels (two-pass, wave32 butterfly reductions)
    float s = a0 + a1 + a2;
    for (int m = 16; m >= 1; m >>= 1) s += __shfl_xor(s, m, 32);
    const float mu = s * (1.0f / 96.0f);
    const float d0 = a0 - mu, d1 = a1 - mu, d2 = a2 - mu;
    float q = d0 * d0 + d1 * d1 + d2 * d2;
    for (int m = 16; m >= 1; m >>= 1) q += __shfl_xor(q, m, 32);
    const float rstd = rsqrtf(q * (1.0f / 96.0f) + kEpsLN);

    float* xo = x_ln + (long)n * kC;
    xo[c0] = d0 * rstd * ln_g[c0] + ln_b[c0];
    xo[c1] = d1 * rstd * ln_g[c1] + ln_b[c1];
    xo[c2] = d2 * rstd * ln_g[c2] + ln_b[c2];
}

// ---------------------------------------------------------------------------
// Kernel 2: pwconv1 (x_ln[N,96] @ w1[96,384] + b1) + exact GELU -> h[N,384]
// WMMA f32 16x16x4. Block = 256 thr = 8 waves, each wave owns 16 rows (M=128
// per block). w1 staged in LDS as 3 panels of 96x128 (stride 136 floats:
// 2*136 mod 64 banks = 16 -> the two half-wave rows of the B operand hit
// disjoint bank sets). A strip (16 rows x 96 K) lives in 48 VGPRs per lane.
// ---------------------------------------------------------------------------
__global__ __launch_bounds__(256) void pw1_kernel(
    const float* __restrict__ x_ln, const float* __restrict__ w1,
    const float* __restrict__ b1, float* __restrict__ hout)
{
    __shared__ float sB[kC * 136];            // 96 rows x 136-float stride

    const int wave = threadIdx.x >> 5;
    const int lane = threadIdx.x & 31;
    const int hl   = lane >> 4;               // half-wave select
    const int col  = lane & 15;
    const int m0   = blockIdx.x * 128 + wave * 16;
    const int row  = m0 + col;

    // Preload the A strip: lane holds x_ln[row][4t+2*hl .. +1] for t=0..23
    float a[48];
    const float* xr = x_ln + (long)row * kC + 2 * hl;
#pragma unroll
    for (int t = 0; t < 24; ++t) {
        v2f v = *(const v2f*)(xr + 4 * t);
        a[2 * t] = v.x; a[2 * t + 1] = v.y;
    }

    for (int nc = 0; nc < 3; ++nc) {          // 3 column panels of 128
        __syncthreads();
        {   // cooperative panel load: 96 x 128 floats as float4s
            const float* gsrc = w1 + nc * 128;
            for (int i = threadIdx.x; i < 96 * 32; i += 256) {
                const int r = i >> 5, c4 = i & 31;
                *(float4*)&sB[r * 136 + c4 * 4] =
                    *(const float4*)(gsrc + (long)r * kH + c4 * 4);
            }
        }
        __syncthreads();

#pragma unroll 1
        for (int nt = 0; nt < 8; ++nt) {      // 8 N-tiles of 16 in this panel
            v8f acc = {0.f, 0.f, 0.f, 0.f, 0.f, 0.f, 0.f, 0.f};
            const float* lb = sB + nt * 16 + col;
#pragma unroll
            for (int t = 0; t < 24; ++t) {    // K = 96 in steps of 4
                v2f av; av.x = a[2 * t]; av.y = a[2 * t + 1];
                v2f bv;
                bv.x = lb[(4 * t + 2 * hl)     * 136];
                bv.y = lb[(4 * t + 2 * hl + 1) * 136];
                acc = __builtin_amdgcn_wmma_f32_16x16x4_f32(
                    false, av, false, bv, (short)0, acc, false, false);
            }
            const int cc = nc * 128 + nt * 16 + col;
            const float bias = b1[cc];
#pragma unroll
            for (int j = 0; j < 8; ++j) {
                float v = acc[j] + bias;
                // exact GELU: x * 0.5 * (1 + erf(x/sqrt(2)))
                v = 0.5f * v * (1.0f + erff(v * 0.70710678118654752440f));
                hout[(long)(m0 + j + 8 * hl) * kH + cc] = v;
            }
        }
    }
}

// ---------------------------------------------------------------------------
// Kernel 3: per-batch segment sum of h*h -> sumsq[B,H].
// batch_idx is sorted, so accumulate locally and flush on batch change.
// 384 threads = one channel each; 256 points per block, fully coalesced.
// ---------------------------------------------------------------------------
__global__ __launch_bounds__(384) void sumsq_kernel(
    const float* __restrict__ h, const int* __restrict__ batch_idx,
    float* __restrict__ sumsq)
{
    const int c  = threadIdx.x;               // channel 0..383
    const int p0 = blockIdx.x * 256;
    int cur = batch_idx[p0];
    float acc = 0.f;
    for (int p = p0; p < p0 + 256; ++p) {
        const int b = batch_idx[p];           // uniform broadcast load
        const float v = h[(long)p * kH + c];
        if (b != cur) {
            atomicAdd(&sumsq[cur * kH + c], acc);
            acc = 0.f; cur = b;
        }
        acc = fmaf(v, v, acc);
    }
    atomicAdd(&sumsq[cur * kH + c], acc);
}

// ---------------------------------------------------------------------------
// Kernel 4: GRN scale s[b,c] = 1 + grn_gamma[c] * Gx/(mean_c(Gx)+eps)
// (the additive grn_beta is folded into cb = grn_beta @ w2 + b2)
// ---------------------------------------------------------------------------
__global__ __launch_bounds__(384) void grn_scale_kernel(
    const float* __restrict__ sumsq, const float* __restrict__ grn_gamma,
    float* __restrict__ s)
{
    __shared__ float red[384];
    const int b = blockIdx.x, c = threadIdx.x;
    const float g = sqrtf(sumsq[b * kH + c]);
    red[c] = g;
    __syncthreads();
    if (c < 128) red[c] += red[c + 256];      // 384 -> 256
    __syncthreads();
    for (int st = 128; st > 0; st >>= 1) {
        if (c < st) red[c] += red[c + st];
        __syncthreads();
    }
    const float mean = red[0] * (1.0f / 384.0f);
    const float nx = g / (mean + kEpsGRN);
    s[b * kH + c] = fmaf(grn_gamma[c], nx, 1.0f);
}

// ---------------------------------------------------------------------------
// Kernel 5: cb[c] = b2[c] + sum_h grn_beta[h] * w2[h,c]   (tiny)
// ---------------------------------------------------------------------------
__global__ __launch_bounds__(96) void cb_kernel(
    const float* __restrict__ grn_beta, const float* __restrict__ w2,
    const float* __restrict__ b2, float* __restrict__ cb)
{
    const int c = threadIdx.x;
    float acc = b2[c];
    for (int hh = 0; hh < kH; ++hh)
        acc = fmaf(grn_beta[hh], w2[hh * kC + c], acc);
    cb[c] = acc;
}

// ---------------------------------------------------------------------------
// Kernel 6: pwconv2 + residual:
//   out = (h .* s[batch]) @ w2 + cb + feats        [N,96]
// WMMA f32 16x16x4, K=384 in 3 LDS panels of 128x96 (stride 104 floats:
// 2*104 mod 64 banks = 16 -> conflict-free half-wave split).
// ---------------------------------------------------------------------------
__global__ __launch_bounds__(256) void pw2_kernel(
    const float* __restrict__ h, const float* __restrict__ s,
    const float* __restrict__ w2, const float* __restrict__ cb,
    const float* __restrict__ feats, const int* __restrict__ batch_idx,
    float* __restrict__ out)
{
    __shared__ float sB[128 * 104];           // 128 K-rows x 104-float stride

    const int wave = threadIdx.x >> 5;
    const int lane = threadIdx.x & 31;
    const int hl   = lane >> 4;
    const int col  = lane & 15;
    const int m0   = blockIdx.x * 128 + wave * 16;
    const int row  = m0 + col;

    const int bidx = batch_idx[row];
    const float* srow = s + bidx * kH;
    const float* hrow = h + (long)row * kH;

    v8f acc[6];
#pragma unroll
    for (int nt = 0; nt < 6; ++nt)
        acc[nt] = (v8f){0.f, 0.f, 0.f, 0.f, 0.f, 0.f, 0.f, 0.f};

    for (int kc = 0; kc < 3; ++kc) {          // 3 K-panels of 128
        __syncthreads();
        {   // cooperative panel load: 128 x 96 floats as float4s
            const float* gsrc = w2 + (long)kc * 128 * kC;
            for (int i = threadIdx.x; i < 128 * 24; i += 256) {
                const int r = i / 24, c4 = i % 24;
                *(float4*)&sB[r * 104 + c4 * 4] =
                    *(const float4*)(gsrc + r * kC + c4 * 4);
            }
        }
        __syncthreads();

#pragma unroll 1
        for (int t = 0; t < 32; ++t) {        // 128 K in steps of 4
            const int kl = 4 * t + 2 * hl;
            const int k0 = kc * 128 + kl;
            v2f av;
            av.x = hrow[k0]     * srow[k0];
            av.y = hrow[k0 + 1] * srow[k0 + 1];
#pragma unroll
            for (int nt = 0; nt < 6; ++nt) {
                v2f bv;
                bv.x = sB[kl       * 104 + nt * 16 + col];
                bv.y = sB[(kl + 1) * 104 + nt * 16 + col];
                acc[nt] = __builtin_amdgcn_wmma_f32_16x16x4_f32(
                    false, av, false, bv, (short)0, acc[nt], false, false);
            }
        }
    }

#pragma unroll
    for (int nt = 0; nt < 6; ++nt) {
        const int cc = nt * 16 + col;
        const float cbv = cb[cc];
#pragma unroll
        for (int j = 0; j < 8; ++j) {
            const long r = m0 + j + 8 * hl;
            out[r * kC + cc] = acc[nt][j] + cbv + feats[r * kC + cc];
        }
    }
}

// ---------------------------------------------------------------------------
// Launch
// ---------------------------------------------------------------------------
extern "C" void kernel_launch(void* const* d_in, const int* in_sizes, int n_in,
                              void* d_out, int out_size, void* d_ws, size_t ws_size,
                              hipStream_t stream)
{
    const float* feats     = (const float*)d_in[0];
    const float* dw_w      = (const float*)d_in[1];
    const float* dw_b      = (const float*)d_in[2];
    const float* ln_gamma  = (const float*)d_in[3];
    const float* ln_beta   = (const float*)d_in[4];
    const float* w1        = (const float*)d_in[5];
    const float* b1        = (const float*)d_in[6];
    const float* grn_gamma = (const float*)d_in[7];
    const float* grn_beta  = (const float*)d_in[8];
    const float* w2        = (const float*)d_in[9];
    const float* b2        = (const float*)d_in[10];
    const int*   nbr       = (const int*)d_in[11];
    const int*   batch_idx = (const int*)d_in[12];
    float*       out       = (float*)d_out;

    // Workspace layout (floats): x_ln[N*C] | h[N*H] | sumsq[B*H] | s[B*H] | cb[C]
    float* ws    = (float*)d_ws;
    float* x_ln  = ws;
    float* hbuf  = x_ln  + (size_t)kN * kC;
    float* sumsq = hbuf  + (size_t)kN * kH;
    float* sbuf  = sumsq + (size_t)kB * kH;
    float* cbuf  = sbuf  + (size_t)kB * kH;

    hipMemsetAsync(sumsq, 0, (size_t)kB * kH * sizeof(float), stream);

    dwconv_ln_kernel<<<kN / 8,   256, 0, stream>>>(feats, dw_w, dw_b, ln_gamma,
                                                   ln_beta, nbr, x_ln);
    pw1_kernel      <<<kN / 128, 256, 0, stream>>>(x_ln, w1, b1, hbuf);
    sumsq_kernel    <<<kN / 256, 384, 0, stream>>>(hbuf, batch_idx, sumsq);
    grn_scale_kernel<<<kB,       384, 0, stream>>>(sumsq, grn_gamma, sbuf);
    cb_kernel       <<<1,         96, 0, stream>>>(grn_beta, w2, b2, cbuf);
    pw2_kernel      <<<kN / 128, 256, 0, stream>>>(hbuf, sbuf, w2, cbuf,
                                                   feats, batch_idx, out);
}